// TSNE_43121471651900
// MI455X (gfx1250) — compile-verified
//
#include <hip/hip_runtime.h>
#include <math.h>

typedef __attribute__((ext_vector_type(2))) float v2f;
typedef __attribute__((ext_vector_type(8))) float v8f;

#define NN 8192
#define CHUNK 1024                 // columns per block
#define TILES (CHUNK / 16)         // 64 WMMA tiles per wave
#define WAVES 8                    // waves per 256-thread block
#define ROWS_PER_BLOCK (WAVES * 16)

// ---------------------------------------------------------------------------
// Zero the three f64 accumulators in workspace (must be re-done every launch).
// ---------------------------------------------------------------------------
__global__ void tsne_init_kernel(double* acc) {
    if (threadIdx.x < 3) acc[threadIdx.x] = 0.0;
}

// ---------------------------------------------------------------------------
// Main streaming pass. One wave = 16 rows; one block = 128 rows x 1024 cols.
// dist tile via V_WMMA_F32_16X16X4_F32 with the rank-4 factorization:
//   a_i = (x0, x1, sq, 1),  b_j = (-2x0, -2x1, 1, sq)  =>  a.b = dist(i,j)
// ---------------------------------------------------------------------------
__global__ __launch_bounds__(256) void tsne_main_kernel(
        const float* __restrict__ x, const float* __restrict__ P,
        double* __restrict__ acc) {
    __shared__ float2 bPair[CHUNK * 2];   // per column: {(-2x0,-2x1), (1, sq)}

    const int tid  = threadIdx.x;
    const int lane = tid & 31;
    const int wave = tid >> 5;
    const int half = lane >> 4;           // 0: lanes 0-15, 1: lanes 16-31
    const int l15  = lane & 15;

    const int rowGroup = blockIdx.x / (NN / CHUNK);
    const int chunk    = blockIdx.x % (NN / CHUNK);
    const int i0    = (rowGroup * WAVES + wave) * 16;  // this wave's 16 rows
    const int jbase = chunk * CHUNK;

    // Stage B operands for the column chunk into LDS (shared by all 8 waves).
    for (int c = tid; c < CHUNK; c += 256) {
        const float2 xc = ((const float2*)x)[jbase + c];
        bPair[c * 2 + 0] = make_float2(-2.0f * xc.x, -2.0f * xc.y);
        bPair[c * 2 + 1] = make_float2(1.0f, xc.x * xc.x + xc.y * xc.y);
    }

    // A operand (16x4 f32): lane l15 = row M; lanes 0-15 hold K=0,1; 16-31 K=2,3.
    const float2 xr = ((const float2*)x)[i0 + l15];
    v2f A;
    if (half == 0) { A[0] = xr.x;                     A[1] = xr.y; }
    else           { A[0] = xr.x*xr.x + xr.y*xr.y;    A[1] = 1.0f; }

    __syncthreads();

    // P pointer matching the WMMA C/D layout: lane L, VGPR e ->
    //   row = i0 + 8*half + e, col = jbase + t*16 + l15  (64B-coalesced b32s)
    const float* prow = P + (size_t)(i0 + 8 * half) * NN + (jbase + l15);

    double a_log = 0.0;   // sum P * log2(P*(1+d))
    double a_p   = 0.0;   // sum P
    double a_q   = 0.0;   // sum 1/(1+d)

    for (int t = 0; t < TILES; ++t) {
        // B operand (4x16 f32): lane l15 = col N; k = 2*half + vgpr
        const float2 bp = bPair[(t * 16 + l15) * 2 + half];
        v2f B; B[0] = bp.x; B[1] = bp.y;

        v8f C = {};
        v8f D = __builtin_amdgcn_wmma_f32_16x16x4_f32(
            /*neg_a=*/false, A, /*neg_b=*/false, B,
            /*c_mod=*/(short)0, C, /*reuse_a=*/false, /*reuse_b=*/false);

        // Non-temporal: P is 268 MB streamed once, keep it out of L2 residency.
        const float* pt = prow + t * 16;
        float p[8];
        #pragma unroll
        for (int e = 0; e < 8; ++e)
            p[e] = __builtin_nontemporal_load(pt + e * NN);

        float lsum = 0.f, psum = 0.f, qsum = 0.f;
        #pragma unroll
        for (int e = 0; e < 8; ++e) {
            const float d  = D[e];
            const float t1 = 1.0f + d;
            qsum += __builtin_amdgcn_rcpf(t1);          // v_rcp_f32
            const float pe = p[e];
            psum += pe;
            const float u = fmaf(pe, d, pe);            // P*(1+d)
            lsum = fmaf(pe, __log2f(u), lsum);          // v_log_f32, fused log
        }
        a_log += (double)lsum;   // f64 accumulate at tile granularity only
        a_p   += (double)psum;
        a_q   += (double)qsum;
    }

    // wave32 reduction
    #pragma unroll
    for (int off = 16; off > 0; off >>= 1) {
        a_log += __shfl_down(a_log, off, 32);
        a_p   += __shfl_down(a_p,   off, 32);
        a_q   += __shfl_down(a_q,   off, 32);
    }
    if (lane == 0) {
        unsafeAtomicAdd(&acc[0], a_log);   // global_atomic_add_f64
        unsafeAtomicAdd(&acc[1], a_p);
        unsafeAtomicAdd(&acc[2], a_q);
    }
}

// ---------------------------------------------------------------------------
// Finalize: loss = ln2 * S_log2 + S_P * ln(S_Q - N)
// ---------------------------------------------------------------------------
__global__ void tsne_final_kernel(const double* __restrict__ acc,
                                  float* __restrict__ out) {
    const double LN2 = 0.69314718055994530941723212145818;
    const double part = acc[2] - (double)NN;
    const double loss = acc[0] * LN2 + acc[1] * log(part);
    out[0] = (float)loss;
}

extern "C" void kernel_launch(void* const* d_in, const int* in_sizes, int n_in,
                              void* d_out, int out_size, void* d_ws, size_t ws_size,
                              hipStream_t stream) {
    const float* x = (const float*)d_in[0];   // [8192, 2] f32
    const float* P = (const float*)d_in[1];   // [8192, 8192] f32
    double* acc = (double*)d_ws;              // 3 f64 accumulators
    float*  out = (float*)d_out;              // scalar f32 loss

    tsne_init_kernel<<<1, 32, 0, stream>>>(acc);

    const int grid = (NN / ROWS_PER_BLOCK) * (NN / CHUNK);   // 64 * 8 = 512
    tsne_main_kernel<<<grid, 256, 0, stream>>>(x, P, acc);

    tsne_final_kernel<<<1, 1, 0, stream>>>(acc, out);
}